// SimpleGrid_2740189135712
// MI455X (gfx1250) — compile-verified
//
#include <hip/hip_runtime.h>
#include <hip/hip_bf16.h>

// Trilinear grid sampling (density + 27-ch SH) at 500k random points.
// Pure random-gather workload: DRAM-bandwidth bound on MI455X (~3.6 GB of
// sector traffic @ 23.3 TB/s ≈ 150 us floor). Strategy:
//  - one point per thread (wave32), 256-thread blocks (8 waves / WGP pass)
//  - paired-z global_load_b64 gathers: 4 requests per channel instead of 8
//  - unroll-3 channel loop -> 12 outstanding B64s/thread (plenty of MLP;
//    the kernel is bandwidth- not latency-bound, so deeper unroll only adds
//    register pressure)
//  - non-temporal (TH=NT) output stores: 56 MB of write-once data stays out
//    of the 192 MB L2, preserving residency for the grids (density plane is
//    fully L2-resident; SH grid gets partial residency).

namespace {

constexpr int RESO = 192;
constexpr int NCH  = 27;                    // BASIS_DIM * 3
constexpr int R2   = RESO * RESO;           // 36864
constexpr int R3   = RESO * RESO * RESO;    // 7077888 (channel stride, elements)

// 8-byte pair with 4-byte alignment: backend emits a single global_load_b64
// for the contiguous (z0, z0+1) pair at any z parity.
struct __attribute__((aligned(4))) F2 { float lo, hi; };

__device__ __forceinline__ F2 ld2(const float* __restrict__ p) {
    return *reinterpret_cast<const F2*>(p);
}

__global__ void __launch_bounds__(256)
trilerp_kernel(const float* __restrict__ pts,   // [N,3]
               const float* __restrict__ dg,    // [1,192,192,192]
               const float* __restrict__ sg,    // [27,192,192,192]
               float* __restrict__ out,         // [N] sigma ++ [N,27] color
               int N)
{
    int n = blockIdx.x * blockDim.x + threadIdx.x;
    if (n >= N) return;

    // Coalesced point read: consecutive threads read consecutive 12B chunks.
    float px = pts[3 * n + 0];
    float py = pts[3 * n + 1];
    float pz = pts[3 * n + 2];

    const float Rm1 = (float)(RESO - 1);
    // Match reference: ((p + 1) * 0.5) * (R - 1), clipped to [0, R-1].
    float ix = fminf(fmaxf((px + 1.0f) * 0.5f * Rm1, 0.0f), Rm1);
    float iy = fminf(fmaxf((py + 1.0f) * 0.5f * Rm1, 0.0f), Rm1);
    float iz = fminf(fmaxf((pz + 1.0f) * 0.5f * Rm1, 0.0f), Rm1);

    // idx >= 0, so int cast == floor; clamp to R-2 so the +1 corner stays in range.
    int x0 = (int)ix; if (x0 > RESO - 2) x0 = RESO - 2;
    int y0 = (int)iy; if (y0 > RESO - 2) y0 = RESO - 2;
    int z0 = (int)iz; if (z0 > RESO - 2) z0 = RESO - 2;

    float fx = ix - (float)x0;
    float fy = iy - (float)y0;
    float fz = iz - (float)z0;
    float gx = 1.0f - fx;
    float gy = 1.0f - fy;
    float gz = 1.0f - fz;

    // xy corner weights; z handled via the paired load (lo = z0, hi = z0+1).
    float w00 = gx * gy;   // (x0, y0)
    float w01 = gx * fy;   // (x0, y1)
    float w10 = fx * gy;   // (x1, y0)
    float w11 = fx * fy;   // (x1, y1)

    // Element index of (x0, y0, z0) within a channel plane. Fits in int (<7.1M).
    int base = (x0 * RESO + y0) * RESO + z0;

    // ---- density (single channel; 28 MB plane stays L2-resident) ----
    {
        const float* g = dg + base;
        F2 a = ld2(g);                 // (x0,y0,z0..z1)
        F2 b = ld2(g + RESO);          // (x0,y1,...)
        F2 c = ld2(g + R2);            // (x1,y0,...)
        F2 d = ld2(g + R2 + RESO);     // (x1,y1,...)
        float nr = w00 * a.lo + w01 * b.lo + w10 * c.lo + w11 * d.lo;
        float fr = w00 * a.hi + w01 * b.hi + w10 * c.hi + w11 * d.hi;
        __builtin_nontemporal_store(gz * nr + fz * fr, out + n);
    }

    // ---- 27 SH channels ----
    float* oc = out + N + (size_t)n * NCH;
    const float* g = sg + base;
#pragma unroll 3
    for (int c = 0; c < NCH; ++c) {
        F2 a  = ld2(g);
        F2 b  = ld2(g + RESO);
        F2 cc = ld2(g + R2);
        F2 d  = ld2(g + R2 + RESO);
        float nr = w00 * a.lo + w01 * b.lo + w10 * cc.lo + w11 * d.lo;
        float fr = w00 * a.hi + w01 * b.hi + w10 * cc.hi + w11 * d.hi;
        __builtin_nontemporal_store(gz * nr + fz * fr, oc + c);
        g += R3;
    }
}

} // anonymous namespace

extern "C" void kernel_launch(void* const* d_in, const int* in_sizes, int n_in,
                              void* d_out, int out_size, void* d_ws, size_t ws_size,
                              hipStream_t stream) {
    const float* pts = (const float*)d_in[0];   // points     [N,3]
    const float* dg  = (const float*)d_in[1];   // density    [1,1,192,192,192]
    const float* sg  = (const float*)d_in[2];   // sh_grid    [1,27,192,192,192]
    float* out = (float*)d_out;                 // [N] sigma ++ [N*27] color

    int N = in_sizes[0] / 3;
    int threads = 256;                          // 8 wave32 waves per block
    int blocks  = (N + threads - 1) / threads;
    trilerp_kernel<<<blocks, threads, 0, stream>>>(pts, dg, sg, out, N);
}